// FlashAttentionImpl_20109036880144
// MI455X (gfx1250) — compile-verified
//
#include <hip/hip_runtime.h>
#include <math.h>

typedef float v2f __attribute__((ext_vector_type(2)));
typedef float v8f __attribute__((ext_vector_type(8)));

#define B_DIM 2
#define H_DIM 16
#define S_LEN 4096
#define D_DIM 64
#define KBLOCK 128                     // matches reference BLOCK_SIZE
#define NKBLK (S_LEN / KBLOCK)         // 32 key blocks
#define QTILES_PER_BH (S_LEN / 16)     // 256 q tiles of 16 rows
#define WAVES_PER_BLOCK 4
#define LDS_STRIDE 138                 // 128 + pad; even -> 8B-aligned b64 LDS loads

// One wave per 16-row Q tile: peak ~230 VGPRs, below the 256-VGPR direct window
// (avoids the s_set_vgpr_msb scalar tax that a 32-row tile incurs on gfx1250).
__global__ __launch_bounds__(WAVES_PER_BLOCK * 32)
void flash_attn_f32_wmma(const float* __restrict__ q,
                         const float* __restrict__ k,
                         const float* __restrict__ v,
                         float* __restrict__ out)
{
    __shared__ float plds[WAVES_PER_BLOCK][16 * LDS_STRIDE];

    const int tid  = threadIdx.x;
    const int wv   = tid >> 5;
    const int lane = tid & 31;
    const int half = lane >> 4;        // 0: lanes 0-15, 1: lanes 16-31
    const int lr   = lane & 15;

    const int wave_global = blockIdx.x * WAVES_PER_BLOCK + wv;
    const int qtile = wave_global & (QTILES_PER_BH - 1);
    const int bh    = wave_global >> 8;

    const float* qbase = q   + ((size_t)bh * S_LEN + (size_t)qtile * 16) * D_DIM;
    const float* kbase = k   + (size_t)bh * S_LEN * D_DIM;
    const float* vbase = v   + (size_t)bh * S_LEN * D_DIM;
    float*       obase = out + ((size_t)bh * S_LEN + (size_t)qtile * 16) * D_DIM;

    // ---- Q as 16 A-fragments (16x4 f32 layout), pre-scaled by 1/sqrt(D)
    v2f qa[16];
#pragma unroll
    for (int g = 0; g < 16; ++g) {
        v2f t = *(const v2f*)(qbase + lr * D_DIM + 4 * g + 2 * half);
        t[0] *= 0.125f; t[1] *= 0.125f;
        qa[g] = t;
    }

    v8f o_acc[4];
#pragma unroll
    for (int t = 0; t < 4; ++t) { v8f z = {}; o_acc[t] = z; }

    float m[8], nrm[8];
#pragma unroll
    for (int r = 0; r < 8; ++r) { m[r] = -INFINITY; nrm[r] = 0.0f; }

    float* pl = &plds[wv][0];

    for (int kb = 0; kb < NKBLK; ++kb) {
        const float* kblk = kbase + (size_t)kb * KBLOCK * D_DIM;
        const float* vblk = vbase + (size_t)kb * KBLOCK * D_DIM;

        if (kb + 1 < NKBLK) {          // warm L2 path for next block
            __builtin_prefetch(kblk + KBLOCK * D_DIM + lane * 256, 0, 0);
            __builtin_prefetch(vblk + KBLOCK * D_DIM + lane * 256, 0, 0);
        }

        // ---- scores: 8 C-tiles of 16x16 covering 128 keys (128 WMMAs)
        v8f c[8];
#pragma unroll
        for (int j = 0; j < 8; ++j) {
            const float* kc_ptr = kblk + j * 16 * D_DIM;
            v2f kbf[16];
#pragma unroll
            for (int g = 0; g < 16; ++g)
                kbf[g] = *(const v2f*)(kc_ptr + lr * D_DIM + 4 * g + 2 * half);
            v8f acc = {};
#pragma unroll
            for (int g = 0; g < 16; ++g)
                acc = __builtin_amdgcn_wmma_f32_16x16x4_f32(
                          false, qa[g], false, kbf[g], (short)0, acc, false, false);
            c[j] = acc;
        }

        // ---- block row-max over all 128 keys (in-register across tiles, then 4 shuffles)
        float bm[8];
#pragma unroll
        for (int r = 0; r < 8; ++r) {
            float x = c[0][r];
#pragma unroll
            for (int j = 1; j < 8; ++j) x = fmaxf(x, c[j][r]);
            x = fmaxf(x, __shfl_xor(x, 1, 32));
            x = fmaxf(x, __shfl_xor(x, 2, 32));
            x = fmaxf(x, __shfl_xor(x, 4, 32));
            x = fmaxf(x, __shfl_xor(x, 8, 32));
            bm[r] = x;
        }

        // ---- reference recurrence: P = exp(S - block_max), carry-scale = exp(m - new_max)
        float esc[8];
#pragma unroll
        for (int r = 0; r < 8; ++r) {
            float nm = fmaxf(m[r], bm[r]);
            esc[r]   = __expf(m[r] - nm);
            m[r]     = nm;
        }

#pragma unroll
        for (int j = 0; j < 8; ++j)
#pragma unroll
            for (int r = 0; r < 8; ++r)
                c[j][r] = __expf(c[j][r] - bm[r]);

#pragma unroll
        for (int r = 0; r < 8; ++r) {
            float s = c[0][r];
#pragma unroll
            for (int j = 1; j < 8; ++j) s += c[j][r];
            s += __shfl_xor(s, 1, 32);
            s += __shfl_xor(s, 2, 32);
            s += __shfl_xor(s, 4, 32);
            s += __shfl_xor(s, 8, 32);
            nrm[r] = nrm[r] * esc[r] + s;
        }

#pragma unroll
        for (int t = 0; t < 4; ++t)
#pragma unroll
            for (int r = 0; r < 8; ++r)
                o_acc[t][r] *= esc[r];

        // ---- transpose P (16x128) from C-layout to A-layout via per-wave LDS
#pragma unroll
        for (int j = 0; j < 8; ++j)
#pragma unroll
            for (int r = 0; r < 8; ++r)
                pl[(r + 8 * half) * LDS_STRIDE + 16 * j + lr] = c[j][r];
        asm volatile("s_wait_dscnt 0" ::: "memory");

        // ---- O += P V : K-depth 128 -> 32 A-fragments x 4 d-tiles (128 WMMAs)
#pragma unroll
        for (int h = 0; h < 32; ++h) {
            v2f pa = *(const v2f*)(pl + lr * LDS_STRIDE + 4 * h + 2 * half);
            const float* v0 = vblk + (4 * h + 2 * half) * D_DIM + lr;
#pragma unroll
            for (int t = 0; t < 4; ++t) {
                v2f vb;
                vb[0] = v0[16 * t];
                vb[1] = v0[16 * t + D_DIM];
                o_acc[t] = __builtin_amdgcn_wmma_f32_16x16x4_f32(
                               false, pa, false, vb, (short)0, o_acc[t], false, false);
            }
        }
    }

    // ---- epilogue: out = O / (norm + 1e-6); C layout: row = r+8*half, col = 16*t + lr
#pragma unroll
    for (int t = 0; t < 4; ++t)
#pragma unroll
        for (int r = 0; r < 8; ++r) {
            float denom = nrm[r] + 1e-6f;
            obase[(r + 8 * half) * D_DIM + 16 * t + lr] = o_acc[t][r] / denom;
        }
}

extern "C" void kernel_launch(void* const* d_in, const int* in_sizes, int n_in,
                              void* d_out, int out_size, void* d_ws, size_t ws_size,
                              hipStream_t stream) {
    const float* q = (const float*)d_in[0];
    const float* k = (const float*)d_in[1];
    const float* v = (const float*)d_in[2];
    float* out = (float*)d_out;

    const int total_waves = B_DIM * H_DIM * QTILES_PER_BH;    // 8192
    dim3 grid(total_waves / WAVES_PER_BLOCK);                  // 2048 blocks
    dim3 block(WAVES_PER_BLOCK * 32);                          // 4 waves (wave32)
    hipLaunchKernelGGL(flash_attn_f32_wmma, grid, block, 0, stream, q, k, v, out);
}